// GranuleCellLayer_87935160418539
// MI455X (gfx1250) — compile-verified
//
#include <hip/hip_runtime.h>

typedef __attribute__((ext_vector_type(2))) float v2f;
typedef __attribute__((ext_vector_type(8))) float v8f;

#define N_MOSSY   1024
#define N_GRANULE 131072
#define TOPK      2621
#define CHUNK     512
#define PAD_STRIDE 516   // 512 + 4 floats pad -> 4-bank shift per row, conflict-free A reads

// ---------------------------------------------------------------------------
// Phase 1: masked matvec  out[r] = sum_k W[r,k]*M[r,k]*mossy[k]
// 16 rows per block, 8 waves, f32 WMMA 16x16x4 accumulation.
// ---------------------------------------------------------------------------
__global__ __launch_bounds__(256) void granule_matvec_kernel(
    const float* __restrict__ mossy,
    const float* __restrict__ W,
    const float* __restrict__ M,
    float* __restrict__ out)
{
    __shared__ float lw[16 * PAD_STRIDE];   // masked weights chunk (33 KB)
    __shared__ float mv[N_MOSSY];           // mossy vector copy   (4 KB)
    __shared__ float part[8][16];           // per-wave partial row sums

    const int t    = threadIdx.x;
    const int lane = t & 31;
    const int w    = t >> 5;          // wave id 0..7
    const int half = lane >> 4;       // 0: K+0,1   1: K+2,3 (A/B fragment halves)
    const int mrow = lane & 15;       // A-matrix row for this lane
    const size_t rowbase = (size_t)blockIdx.x * 16;

    // stage mossy vector: 1024 floats = 256 float4, one per thread
    {
        const float4* m4 = (const float4*)mossy;
        ((float4*)mv)[t] = m4[t];
    }

    v8f c = {};  // 16x16 f32 accumulator tile (column 0 holds the dot products)

    for (int ch = 0; ch < 2; ++ch) {
        __syncthreads();
        // coalesced load of 16 x 512 masked weights into LDS
        const size_t colbase = (size_t)ch * CHUNK;
        for (int i = 0; i < 8; ++i) {
            int g   = t + i * 256;      // float4 index 0..2047
            int row = g >> 7;           // 128 float4 per row
            int c4  = g & 127;
            size_t gidx = (rowbase + (size_t)row) * N_MOSSY + colbase + (size_t)c4 * 4;
            float4 wv = *(const float4*)(W + gidx);
            float4 mk = *(const float4*)(M + gidx);
            float* dst = &lw[row * PAD_STRIDE + c4 * 4];
            dst[0] = wv.x * mk.x; dst[1] = wv.y * mk.y;
            dst[2] = wv.z * mk.z; dst[3] = wv.w * mk.w;
        }
        __syncthreads();

        // this wave's 64-wide K slice within the chunk: 16 WMMA steps of K=4
        const int kw = w * 64;
        for (int s = 0; s < 16; ++s) {
            const int kl = kw + s * 4;             // local k in chunk
            const int kg = ch * CHUNK + kl;        // global k
            // A fragment: lane half 0 -> (K+0,K+1), half 1 -> (K+2,K+3), row = mrow
            v2f a = *(const v2f*)&lw[mrow * PAD_STRIDE + kl + 2 * half];
            // B fragment: mossy replicated across all 16 columns
            v2f b = *(const v2f*)&mv[kg + 2 * half];
            c = __builtin_amdgcn_wmma_f32_16x16x4_f32(
                    false, a, false, b, (short)0, c, false, false);
        }
    }

    // column 0 of C: lane 0 holds rows 0..7 (v0..v7), lane 16 holds rows 8..15
    if (mrow == 0) {
        float* p = &part[w][half * 8];
        p[0] = c[0]; p[1] = c[1]; p[2] = c[2]; p[3] = c[3];
        p[4] = c[4]; p[5] = c[5]; p[6] = c[6]; p[7] = c[7];
    }
    __syncthreads();

    if (t < 16) {
        float s = 0.0f;
        for (int j = 0; j < 8; ++j) s += part[j][t];
        out[rowbase + t] = s;
    }
}

// ---------------------------------------------------------------------------
// Phase 2: exact radix-select of the K-th largest value (monotonic uint keys)
// ---------------------------------------------------------------------------
__device__ __forceinline__ unsigned keyOf(float f) {
    unsigned u = __float_as_uint(f);
    return (u & 0x80000000u) ? ~u : (u | 0x80000000u);
}

__global__ void select_init_kernel(unsigned* hist, unsigned* state, unsigned kval) {
    hist[threadIdx.x] = 0u;
    if (threadIdx.x == 0) { state[0] = 0u; state[1] = kval; }
}

__global__ __launch_bounds__(256) void select_hist_kernel(
    const float* __restrict__ vals, unsigned* __restrict__ hist,
    const unsigned* __restrict__ state, unsigned shift, unsigned highmask, int n)
{
    __shared__ unsigned sh[256];
    sh[threadIdx.x] = 0u;
    __syncthreads();
    const unsigned prefix = state[0];
    const int stride = gridDim.x * blockDim.x;
    for (int i = blockIdx.x * blockDim.x + threadIdx.x; i < n; i += stride) {
        unsigned k = keyOf(vals[i]);
        if ((k & highmask) == prefix)
            atomicAdd(&sh[(k >> shift) & 255u], 1u);
    }
    __syncthreads();
    unsigned v = sh[threadIdx.x];
    if (v) atomicAdd(&hist[threadIdx.x], v);
}

__global__ void select_scan_kernel(unsigned* hist, unsigned* state, unsigned shift) {
    __shared__ unsigned sh[256];
    sh[threadIdx.x] = hist[threadIdx.x];
    __syncthreads();
    if (threadIdx.x == 0) {
        unsigned kRem = state[1];
        unsigned acc = 0, b = 0;
        for (int j = 255; j >= 0; --j) {
            unsigned cnt = sh[j];
            if (acc + cnt >= kRem) { b = (unsigned)j; kRem -= acc; break; }
            acc += cnt;
        }
        state[1] = kRem;
        state[0] |= b << shift;
    }
    hist[threadIdx.x] = 0u;   // ready for next pass
}

__global__ __launch_bounds__(256) void select_apply_kernel(
    float* __restrict__ out, const unsigned* __restrict__ state,
    const float* __restrict__ thr, int n)
{
    int i = blockIdx.x * blockDim.x + threadIdx.x;
    if (i < n) {
        float v = out[i];
        unsigned k = keyOf(v);
        float th = *thr;
        out[i] = (k >= state[0]) ? fmaxf(v - th, 0.0f) : 0.0f;
    }
}

// ---------------------------------------------------------------------------
extern "C" void kernel_launch(void* const* d_in, const int* in_sizes, int n_in,
                              void* d_out, int out_size, void* d_ws, size_t ws_size,
                              hipStream_t stream)
{
    (void)in_sizes; (void)n_in; (void)out_size; (void)ws_size;
    const float* mossy = (const float*)d_in[0];   // [1024]
    const float* W     = (const float*)d_in[1];   // [131072,1024]
    const float* thr   = (const float*)d_in[2];   // scalar
    const float* Mmask = (const float*)d_in[3];   // [131072,1024]
    float* out         = (float*)d_out;           // [131072]

    unsigned* hist  = (unsigned*)d_ws;            // 256 buckets
    unsigned* state = hist + 256;                 // [0]=prefix key, [1]=k remaining

    // Phase 1: masked matvec (HBM-bound, ~1.07 GB streamed)
    granule_matvec_kernel<<<N_GRANULE / 16, 256, 0, stream>>>(mossy, W, Mmask, out);

    // Phase 2: exact top-K threshold via 4-pass radix select over 32-bit keys
    select_init_kernel<<<1, 256, 0, stream>>>(hist, state, (unsigned)TOPK);
    for (int p = 0; p < 4; ++p) {
        unsigned shift = 24u - 8u * (unsigned)p;
        unsigned highmask = (p == 0) ? 0u : (0xFFFFFFFFu << (shift + 8u));
        select_hist_kernel<<<64, 256, 0, stream>>>(out, hist, state, shift, highmask, N_GRANULE);
        select_scan_kernel<<<1, 256, 0, stream>>>(hist, state, shift);
    }
    select_apply_kernel<<<(N_GRANULE + 255) / 256, 256, 0, stream>>>(out, state, thr, N_GRANULE);
}